// GNNLayer_43671227466246
// MI455X (gfx1250) — compile-verified
//
#include <hip/hip_runtime.h>

// GCNConv (no self loops) -> ReLU -> residual, for MI455X (gfx1250, wave32).
//   x:          [50000, 128] f32
//   edge_index: [2, 600000]  int32 (harness casts integers to int*)
//   W:          [128, 128]   f32
//   b:          [128]        f32
// out = relu(scatter_add(norm * (x@W^T)[row] -> col) + b) + x

#define N_NODES 50000
#define N_EDGES 600000
#define DIM 128
#define MTILES (N_NODES / 16)   // 3125, exact

typedef __attribute__((ext_vector_type(2))) float v2f;
typedef __attribute__((ext_vector_type(8))) float v8f;

// ---------------------------------------------------------------- degree ----
__global__ __launch_bounds__(256) void deg_kernel(const int* __restrict__ ei,
                                                  float* __restrict__ deg) {
    int e = blockIdx.x * blockDim.x + threadIdx.x;
    if (e < N_EDGES) {
        int c = ei[N_EDGES + e];          // col = edge_index[1][e]
        unsafeAtomicAdd(&deg[c], 1.0f);   // global_atomic_add_f32
    }
}

// ------------------------------------------------------------- deg^-1/2 ----
__global__ __launch_bounds__(256) void dinv_kernel(float* __restrict__ deg) {
    int i = blockIdx.x * blockDim.x + threadIdx.x;
    if (i < N_NODES) {
        float d = deg[i];
        deg[i] = (d > 0.0f) ? __frsqrt_rn(d) : 0.0f;
    }
}

// ------------------------------------------------- x_lin = x @ W^T (WMMA) ----
// One block: 16 rows x 128 cols of x_lin. 8 waves, wave w owns N-tile w.
// FP32 WMMA 16x16x4, K = 128 -> 32 WMMA ops per wave, fully unrolled.
//
// A (16x4, f32, 2 VGPRs/lane): lanes 0-15 hold M=lane {K=k0, k0+1},
//                              lanes 16-31 hold M=lane-16 {K=k0+2, k0+3}.
// B (4x16, f32, 2 VGPRs/lane): lanes 0-15 hold N=lane {K=k0, k0+1},
//                              lanes 16-31 hold N=lane-16 {K=k0+2, k0+3}.
//   B[k][n] = W[n][k]  (since x_lin = x @ W^T).
// C/D (16x16 f32, 8 VGPRs): c[v] = element [M = v + 8*(lane>=16)][N = lane&15].
__global__ __launch_bounds__(256) void gemm_wmma_kernel(const float* __restrict__ x,
                                                        const float* __restrict__ W,
                                                        float* __restrict__ xlin) {
    const int wave = threadIdx.x >> 5;        // 0..7 -> N tile
    const int lane = threadIdx.x & 31;
    const int half = lane >> 4;               // 0 or 1
    const int l16  = lane & 15;
    const int m0   = blockIdx.x * 16;         // row-tile base
    const int n0   = wave * 16;               // col-tile base

    const float* arow = x + (size_t)(m0 + l16) * DIM;          // A: row m0+l16
    const float* brow = W + (size_t)(n0 + l16) * DIM;          // B: W row n0+l16

    v8f c = {};
#pragma unroll
    for (int k0 = 0; k0 < DIM; k0 += 4) {
        v2f a, bm;
        a.x  = arow[k0 + 2 * half];
        a.y  = arow[k0 + 2 * half + 1];
        bm.x = brow[k0 + 2 * half];
        bm.y = brow[k0 + 2 * half + 1];
        // 8 args: (neg_a, A, neg_b, B, c_mod, C, reuse_a, reuse_b)
        c = __builtin_amdgcn_wmma_f32_16x16x4_f32(false, a, false, bm,
                                                  (short)0, c, false, false);
    }

    float* orow = xlin + (size_t)(m0 + 8 * half) * DIM + n0 + l16;
#pragma unroll
    for (int v = 0; v < 8; ++v)
        orow[(size_t)v * DIM] = c[v];
}

// ------------------------------------- gather + edge-norm + scatter-add ----
// 32 threads per edge, float4 per thread: xlin row gathered as 32 x b128
// loads (whole 512B row, cacheline-coalesced), 4 HW f32 atomics per thread.
__global__ __launch_bounds__(256) void scatter_kernel(const int* __restrict__ ei,
                                                      const float* __restrict__ dinv,
                                                      const float* __restrict__ xlin,
                                                      float* __restrict__ agg) {
    long long idx = (long long)blockIdx.x * blockDim.x + threadIdx.x;
    int e = (int)(idx >> 5);           // edge
    int q = (int)(idx & 31);           // float4 slot within the 128-wide row
    if (e < N_EDGES) {
        int r = ei[e];                 // source
        int c = ei[N_EDGES + e];       // target
        float norm = dinv[r] * dinv[c];
        const float4 v = ((const float4*)(xlin + (size_t)r * DIM))[q];
        float* dst = agg + (size_t)c * DIM + q * 4;
        unsafeAtomicAdd(dst + 0, v.x * norm);
        unsafeAtomicAdd(dst + 1, v.y * norm);
        unsafeAtomicAdd(dst + 2, v.z * norm);
        unsafeAtomicAdd(dst + 3, v.w * norm);
    }
}

// ------------------------------------------- relu(agg + b) + x epilogue ----
__global__ __launch_bounds__(256) void epilogue_kernel(const float* __restrict__ x,
                                                       const float* __restrict__ b,
                                                       float* __restrict__ out) {
    int i = blockIdx.x * blockDim.x + threadIdx.x;   // float4 index
    if (i < N_NODES * DIM / 4) {
        float4 acc = ((float4*)out)[i];
        float4 xb  = ((const float4*)x)[i];
        float4 bb  = ((const float4*)b)[i & 31];
        float4 r;
        r.x = fmaxf(acc.x + bb.x, 0.0f) + xb.x;
        r.y = fmaxf(acc.y + bb.y, 0.0f) + xb.y;
        r.z = fmaxf(acc.z + bb.z, 0.0f) + xb.z;
        r.w = fmaxf(acc.w + bb.w, 0.0f) + xb.w;
        ((float4*)out)[i] = r;
    }
}

// ---------------------------------------------------------------------------
extern "C" void kernel_launch(void* const* d_in, const int* in_sizes, int n_in,
                              void* d_out, int out_size, void* d_ws, size_t ws_size,
                              hipStream_t stream) {
    const float* x  = (const float*)d_in[0];
    const int*   ei = (const int*)d_in[1];
    const float* W  = (const float*)d_in[2];
    const float* b  = (const float*)d_in[3];
    float*       out = (float*)d_out;

    float* deg  = (float*)d_ws;          // N_NODES floats (becomes deg^-1/2)
    float* xlin = deg + N_NODES;         // N_NODES*DIM floats

    // Accumulator (d_out) and degree buffer must be zeroed EVERY call.
    hipMemsetAsync(deg, 0, (size_t)N_NODES * sizeof(float), stream);
    hipMemsetAsync(out, 0, (size_t)N_NODES * DIM * sizeof(float), stream);

    deg_kernel<<<(N_EDGES + 255) / 256, 256, 0, stream>>>(ei, deg);
    dinv_kernel<<<(N_NODES + 255) / 256, 256, 0, stream>>>(deg);
    gemm_wmma_kernel<<<MTILES, 256, 0, stream>>>(x, W, xlin);

    long long sc_threads = (long long)N_EDGES * 32;
    int sc_blocks = (int)((sc_threads + 255) / 256);
    scatter_kernel<<<sc_blocks, 256, 0, stream>>>(ei, deg, xlin, out);

    int ep_threads = N_NODES * DIM / 4;
    epilogue_kernel<<<(ep_threads + 255) / 256, 256, 0, stream>>>(x, b, out);
}